// PositionalEncoding_833223655827
// MI455X (gfx1250) — compile-verified
//
#include <hip/hip_runtime.h>

// Positional-encoding add (B=16, S=4096, D=512, PE_DIM=256).
// HBM-bound elementwise op (~270 MB moved -> ~12 us roofline @ 23.3 TB/s);
// no contraction dim, so WMMA is not applicable. MI455X tuning:
//  - b128 loads/stores throughout
//  - each thread produces BOTH d-halves (d, d+256) from ONE pair of pe quads
//    (the tile(1,2) structure makes them identical) -> halves pe traffic + math
//  - nontemporal hints on streaming x/out so the 5 MB pe table stays cached
//  - gfx1250 global_prefetch_b8 read-ahead
//  - s (and valid / rev_row) is wave-uniform -> scalarizes under wave32

typedef __attribute__((ext_vector_type(4))) float v4f;

#define S_LEN   4096
#define DMODEL  512
#define PEDIM   256
#define HALF    256    // DMODEL / 2
#define TPB     256
#define VPT     4      // column-group pairs per thread

__global__ __launch_bounds__(TPB)
void pe_lengths_kernel(const int* __restrict__ mask, int* __restrict__ lengths)
{
    const int b = blockIdx.x;
    const int* m = mask + (size_t)b * S_LEN;
    int cnt = 0;
    for (int i = threadIdx.x; i < S_LEN; i += TPB)
        cnt += (m[i] == 0) ? 1 : 0;

    __shared__ int red[TPB];
    red[threadIdx.x] = cnt;
    __syncthreads();
    #pragma unroll
    for (int off = TPB / 2; off > 0; off >>= 1) {
        if (threadIdx.x < off) red[threadIdx.x] += red[threadIdx.x + off];
        __syncthreads();
    }
    if (threadIdx.x == 0) lengths[b] = red[0];
}

__global__ __launch_bounds__(TPB)
void pe_add_kernel(const float* __restrict__ x,
                   const float* __restrict__ pe,
                   const int*   __restrict__ lengths,
                   float*       __restrict__ out)
{
    const int b = blockIdx.y;
    const int L = lengths[b];                       // wave-uniform scalar load
    const size_t boff = (size_t)b * S_LEN * DMODEL;
    const float* xb = x + boff;
    float*       ob = out + boff;

    // n indexes (row, column-group): n = s*64 + c4, c4 in [0,64)
    const int n_block = blockIdx.x * (TPB * VPT);

    // Read-ahead of the next block's x region (gfx1250 global_prefetch_b8).
    {
        const int np = n_block + TPB * VPT;
        const size_t pf = (size_t)(np >> 6) * DMODEL + (size_t)(np & 63) * 4;
        __builtin_prefetch(xb + pf + threadIdx.x * 4, 0, 1);
    }

    const v4f zero = {0.f, 0.f, 0.f, 0.f};

    #pragma unroll
    for (int i = 0; i < VPT; ++i) {
        const int n  = n_block + i * TPB + (int)threadIdx.x;
        const int s  = n >> 6;            // 64 column-groups per row
        const int c0 = (n & 63) << 2;     // pe column start, 16B aligned

        const size_t row = (size_t)s * DMODEL;

        // Streaming x loads for both halves (non-temporal: read-once data).
        const v4f xv0 = __builtin_nontemporal_load((const v4f*)(xb + row + c0));
        const v4f xv1 = __builtin_nontemporal_load((const v4f*)(xb + row + HALF + c0));

        // Base PE quad (cached; reused across batches and both halves).
        const v4f bp = *(const v4f*)(pe + (size_t)s * PEDIM + c0);

        // Reversed-feature PE: columns [255-c0 .. 252-c0] = aligned quad at 252-c0,
        // consumed in reversed component order. Shared by both halves.
        const bool valid   = (s < L);                  // wave-uniform
        const int  rev_row = valid ? (L - 1 - s) : 0;  // in [0, 4095] when valid
        const v4f  rr  = *(const v4f*)(pe + (size_t)rev_row * PEDIM + (252 - c0));
        v4f rev = rr.wzyx;
        rev = valid ? rev : zero;

        // Match reference order: (x + add_rev) + base.
        const v4f o0 = (xv0 + rev) + bp;
        const v4f o1 = (xv1 + rev) + bp;

        __builtin_nontemporal_store(o0, (v4f*)(ob + row + c0));
        __builtin_nontemporal_store(o1, (v4f*)(ob + row + HALF + c0));
    }
}

extern "C" void kernel_launch(void* const* d_in, const int* in_sizes, int n_in,
                              void* d_out, int out_size, void* d_ws, size_t ws_size,
                              hipStream_t stream)
{
    const float* x    = (const float*)d_in[0];  // (16, 4096, 512) f32
    const int*   mask = (const int*)d_in[1];    // (16, 4096) i32
    const float* pe   = (const float*)d_in[2];  // (5000, 256) f32
    float*       out  = (float*)d_out;          // (16, 4096, 512) f32
    int*         lengths = (int*)d_ws;          // 16 ints of scratch

    pe_lengths_kernel<<<dim3(16), dim3(TPB), 0, stream>>>(mask, lengths);

    // Per batch: S * 64 = 262144 column-group pairs; per block: TPB*VPT = 1024
    // -> 256 blocks per batch.
    dim3 grid(256, 16);
    pe_add_kernel<<<grid, dim3(TPB), 0, stream>>>(x, pe, lengths, out);
}